// SecureOptimizedBlockReLU_21122649161900
// MI455X (gfx1250) — compile-verified
//
#include <hip/hip_runtime.h>
#include <hip/hip_bf16.h>
#include <stdint.h>

#define N_  16
#define C_  128
#define H_  114
#define W_  114
#define P_  (H_*W_)        /* 12996 floats per plane; plane stride = 51984 B (16B aligned) */
#define NCA 32             /* passthrough channels [0,32)  */
#define NCB 48             /* 2x2 channels       [32,80)  */
#define NCC 48             /* 4x4 channels       [80,128) */

typedef int v2i __attribute__((ext_vector_type(2)));
typedef __attribute__((address_space(1))) v2i Gv2i;   /* global int2 */
typedef __attribute__((address_space(3))) v2i Lv2i;   /* LDS    int2 */

#if defined(__gfx1250__) && \
    __has_builtin(__builtin_amdgcn_global_load_async_to_lds_b64) && \
    __has_builtin(__builtin_amdgcn_s_wait_asynccnt)
#define USE_ASYNC_LDS 1
#else
#define USE_ASYNC_LDS 0
#endif

// ---------------------------------------------------------------------------
// Channels [0,32): pure passthrough. Bandwidth-bound memcpy of 26.6 MB.
// CDNA5 path: stream global -> LDS via async DMA (ASYNCcnt), then LDS -> out.
// Fallback: direct float4 streaming with global_prefetch_b8 lookahead.
// ---------------------------------------------------------------------------
__global__ void __launch_bounds__(256) copy_pass(const float* __restrict__ src,
                                                 float* __restrict__ dst) {
#if USE_ASYNC_LDS
  __shared__ float2 lbuf[256];
  const size_t nf2_per_n = (size_t)NCA * P_ / 2;        // 207936 float2 per image
  const size_t total     = (size_t)N_ * nf2_per_n;      // 3,326,976
  const size_t stride    = (size_t)gridDim.x * blockDim.x;
  // AS3 pointer value == LDS byte offset == low 32 bits of the generic address
  const uint32_t lds_off = (uint32_t)(uintptr_t)&lbuf[threadIdx.x];
  for (size_t i = (size_t)blockIdx.x * blockDim.x + threadIdx.x; i < total; i += stride) {
    size_t n   = i / nf2_per_n;
    size_t r   = i - n * nf2_per_n;
    size_t off = n * ((size_t)C_ * P_ / 2) + r;         // float2 units
    const float2* g = (const float2*)src + off;
    __builtin_prefetch(g + stride, 0, 0);               // global_prefetch_b8
    // global_load_async_to_lds_b64: lane-private 8B slot, tracked by ASYNCcnt
    __builtin_amdgcn_global_load_async_to_lds_b64(
        (Gv2i*)(uintptr_t)g, (Lv2i*)lds_off, 0, 0);
    __builtin_amdgcn_s_wait_asynccnt(0);                // s_wait_asynccnt 0
    asm volatile("" ::: "memory");                      // keep the LDS read below the wait
    float2 v = lbuf[threadIdx.x];
    ((float2*)dst)[off] = v;
  }
#else
  const size_t nf4_per_n = (size_t)NCA * P_ / 4;        // 103968 float4 per image
  const size_t total     = (size_t)N_ * nf4_per_n;      // 1,663,488
  const size_t stride    = (size_t)gridDim.x * blockDim.x;
  for (size_t i = (size_t)blockIdx.x * blockDim.x + threadIdx.x; i < total; i += stride) {
    size_t n   = i / nf4_per_n;
    size_t r   = i - n * nf4_per_n;
    size_t off = n * (size_t)(C_ * P_) + r * 4;
    const float4* g = (const float4*)(src + off);
    __builtin_prefetch((const char*)g + 65536, 0, 0);   // global_prefetch_b8
    *(float4*)(dst + off) = *g;
  }
#endif
}

// ---------------------------------------------------------------------------
// Channels [32,80): 2x2 blocks. One thread per block: two 8B-aligned float2
// loads (rows 2by, 2by+1), sum, keep iff bit 22 of int64(sum) is clear.
// 57*57 blocks/plane; consecutive lanes -> consecutive 8B -> fully coalesced.
// ---------------------------------------------------------------------------
__global__ void __launch_bounds__(256) mask2x2(const float* __restrict__ src,
                                               float* __restrict__ dst) {
  const int BX = W_ / 2;                 // 57
  const int BY = H_ / 2;                 // 57
  const int total = N_ * NCB * BY * BX;  // 2,495,232 == 9747 * 256
  int t = blockIdx.x * blockDim.x + threadIdx.x;
  if (t >= total) return;
  int bx = t % BX;
  int t2 = t / BX;
  int by = t2 % BY;
  int pc = t2 / BY;                      // 0..N_*NCB-1
  int n  = pc / NCB;
  int c  = NCA + (pc - n * NCB);
  size_t base = ((size_t)n * C_ + c) * P_ + (size_t)(2 * by) * W_ + 2 * bx;
  float2 a = *(const float2*)(src + base);
  float2 b = *(const float2*)(src + base + W_);
  float s = (a.x + a.y) + (b.x + b.y);
  long long si = (long long)s;           // trunc toward zero == astype(int64)
  float m = ((si >> 22) & 1LL) ? 0.0f : 1.0f;  // bit 20 of (s>>2) == bit 22 of s
  *(float2*)(dst + base)      = make_float2(a.x * m, a.y * m);
  *(float2*)(dst + base + W_) = make_float2(b.x * m, b.y * m);
}

// ---------------------------------------------------------------------------
// Channels [80,128): 4x4 blocks with zero padding to 116x116. One thread per
// block: up to 4 rows x two float2 loads (all 8B aligned). Ragged edges at
// bx==28 (cols 112..113 valid) and by==28 (rows 112..113 valid); padding
// contributes zero to the sum and is never written. Keep iff bit 24 clear.
// ---------------------------------------------------------------------------
__global__ void __launch_bounds__(256) mask4x4(const float* __restrict__ src,
                                               float* __restrict__ dst) {
  const int BX = (W_ + 3) / 4;           // 29
  const int BY = (H_ + 3) / 4;           // 29
  const int total = N_ * NCC * BY * BX;  // 645,888 == 2523 * 256
  int t = blockIdx.x * blockDim.x + threadIdx.x;
  if (t >= total) return;
  int bx = t % BX;
  int t2 = t / BX;
  int by = t2 % BY;
  int pc = t2 / BY;
  int n  = pc / NCC;
  int c  = (NCA + NCB) + (pc - n * NCC);
  int x0 = 4 * bx, y0 = 4 * by;
  int ny = (y0 + 4 <= H_) ? 4 : (H_ - y0);      // 4, or 2 at by==28
  bool full_x = (x0 + 4 <= W_);                 // false only at bx==28
  size_t base = ((size_t)n * C_ + c) * P_;

  float2 v0[4], v1[4];
  float s = 0.0f;
#pragma unroll
  for (int i = 0; i < 4; ++i) {
    if (i < ny) {
      const float* rp = src + base + (size_t)(y0 + i) * W_ + x0;
      v0[i] = *(const float2*)rp;
      s += v0[i].x + v0[i].y;
      if (full_x) {
        v1[i] = *(const float2*)(rp + 2);
        s += v1[i].x + v1[i].y;
      }
    }
  }
  long long si = (long long)s;
  float m = ((si >> 24) & 1LL) ? 0.0f : 1.0f;   // bit 20 of (s>>4) == bit 24 of s
#pragma unroll
  for (int i = 0; i < 4; ++i) {
    if (i < ny) {
      float* rp = dst + base + (size_t)(y0 + i) * W_ + x0;
      *(float2*)rp = make_float2(v0[i].x * m, v0[i].y * m);
      if (full_x) *(float2*)(rp + 2) = make_float2(v1[i].x * m, v1[i].y * m);
    }
  }
}

extern "C" void kernel_launch(void* const* d_in, const int* in_sizes, int n_in,
                              void* d_out, int out_size, void* d_ws, size_t ws_size,
                              hipStream_t stream) {
  (void)in_sizes; (void)n_in; (void)out_size; (void)d_ws; (void)ws_size;
  const float* act = (const float*)d_in[0];
  float* out = (float*)d_out;

  // Passthrough copy, channels [0,32): grid-stride streaming.
  copy_pass<<<4096, 256, 0, stream>>>(act, out);

  // 2x2 masking, channels [32,80): one thread per block, exact grid.
  const int tB = N_ * NCB * (H_ / 2) * (W_ / 2);        // 2,495,232
  mask2x2<<<(tB + 255) / 256, 256, 0, stream>>>(act, out);

  // 4x4 masking, channels [80,128): one thread per block, exact grid.
  const int tC = N_ * NCC * ((H_ + 3) / 4) * ((W_ + 3) / 4);  // 645,888
  mask4x4<<<(tC + 255) / 256, 256, 0, stream>>>(act, out);
}